// jslab_kt_Ue_Unio_25881472925758
// MI455X (gfx1250) — compile-verified
//
#include <hip/hip_runtime.h>
#include <math.h>

// ---------------- model constants (match reference) ----------------
#define NTT   2160          // NT == NFORCING
#define NDTK  9             // basis knots
#define DTF   60.0f         // Euler dt
#define FCF   1e-4f         // Coriolis
#define DTKF  864000.0f     // knot spacing
#define DTHF  3600.0f       // forcing dt
#define NSUBI 60

typedef __attribute__((ext_vector_type(2))) float v2f;
typedef __attribute__((ext_vector_type(8))) float v8f;

// Gaussian basis weight (un-normalized) for forcing-step m, knot j
__device__ __forceinline__ float gaussw(int m, int j) {
    float t = (float)m * DTHF;
    float d = (t - (float)j * DTKF) * (1.0f / DTKF);
    return expf(-0.5f * d * d);
}

// ---------------- Kernel 1: Kt = M @ exp(pk).reshape(9,2) via WMMA ----------------
// One wave (32 threads) per 16-row tile of Kt; 135 tiles. K-dim padded 9->12,
// N-dim padded 2->16 (zero columns). 3 x V_WMMA_F32_16X16X4_F32 per tile.
__global__ void __launch_bounds__(32)
k_kt_wmma(const float* __restrict__ pk, float* __restrict__ Kt0, float* __restrict__ Kt1) {
    const int tile    = blockIdx.x;        // 0..134
    const int lane    = threadIdx.x;       // 0..31
    const int halfsel = lane >> 4;         // 0: lanes 0-15, 1: lanes 16-31
    const int col     = lane & 15;
    const int m       = tile * 16 + col;   // this lane's A-row (for A operand)

    // row-normalization factor for row m of M
    float s = 0.0f;
#pragma unroll
    for (int j = 0; j < NDTK; ++j) s += gaussw(m, j);
    const float invs = 1.0f / s;

    v8f acc = {};
#pragma unroll
    for (int c = 0; c < 3; ++c) {
        // A (16x4 f32): lane half picks K = {0,1} or {2,3} within this 4-slice
        const int k0 = 4 * c + 2 * halfsel;
        v2f a;
        a.x = (k0     < NDTK) ? gaussw(m, k0    ) * invs : 0.0f;
        a.y = (k0 + 1 < NDTK) ? gaussw(m, k0 + 1) * invs : 0.0f;
        // B (4x16 f32): VGPR v holds row K=v (lanes 0-15) / K=v+2 (lanes 16-31)
        const int r0 = 4 * c + 2 * halfsel;
        v2f b;
        b.x = (r0     < NDTK && col < 2) ? expf(pk[ r0      * 2 + col]) : 0.0f;
        b.y = (r0 + 1 < NDTK && col < 2) ? expf(pk[(r0 + 1) * 2 + col]) : 0.0f;
        acc = __builtin_amdgcn_wmma_f32_16x16x4_f32(
                  false, a, false, b, (short)0, acc, false, false);
    }

    // D layout: VGPR r -> row r (lanes 0-15) / row r+8 (lanes 16-31), col = lane&15.
    if (col < 2) {
#pragma unroll
        for (int r = 0; r < 8; ++r) {
            const int row = tile * 16 + r + 8 * halfsel;
            const float v = acc[r];
            if (col == 0) Kt0[row] = v; else Kt1[row] = v;
        }
    }
}

// ---------------- Kernel 2: Ekman current Ue,Ve ----------------
__global__ void k_ekman(const float* __restrict__ Kt0, const float* __restrict__ Kt1,
                        const float* __restrict__ TAx, const float* __restrict__ TAy,
                        float* __restrict__ Ue, float* __restrict__ Ve) {
    const int t = blockIdx.x * blockDim.x + threadIdx.x;
    if (t >= NTT) return;
    const float k0 = Kt0[t], k1 = Kt1[t];
    const float tx = TAx[t], ty = TAy[t];
    const float inv = 1.0f / (k1 * k1 + FCF * FCF);
    Ue[t] = k0 * (tx * k1 + ty * FCF) * inv;
    Ve[t] = k0 * (ty * k1 - tx * FCF) * inv;
}

// ---------------- Kernel 3: dCe/dt via jnp.gradient with spacing DT=60 ----------------
__global__ void k_grad(const float* __restrict__ Ue, const float* __restrict__ Ve,
                       float* __restrict__ dUe, float* __restrict__ dVe) {
    const int t = blockIdx.x * blockDim.x + threadIdx.x;
    if (t >= NTT) return;
    float du, dv;
    if (t == 0) {
        du = (Ue[1] - Ue[0]) / DTF;          dv = (Ve[1] - Ve[0]) / DTF;
    } else if (t == NTT - 1) {
        du = (Ue[NTT-1] - Ue[NTT-2]) / DTF;  dv = (Ve[NTT-1] - Ve[NTT-2]) / DTF;
    } else {
        du = (Ue[t+1] - Ue[t-1]) / (2.0f*DTF); dv = (Ve[t+1] - Ve[t-1]) / (2.0f*DTF);
    }
    dUe[t] = du; dVe[t] = dv;
}

// ---------------- Kernel 4: per-hour affine maps (60 Euler substeps composed) ----
// C' = a*C + b with a = 1 - DT*Ktnow1 - i*DT*FC, b = -DT*(dUe + i*dVe).
// Reproduces reference indexing: Kt[it-1] wraps(-1->2159)/clamps(>=NT->2159);
// itsup = -1 -> 2159 when iout+1 >= NT.
__global__ void k_chunks(const float* __restrict__ Kt1,
                         const float* __restrict__ dUe, const float* __restrict__ dVe,
                         float* __restrict__ Are, float* __restrict__ Aim,
                         float* __restrict__ Bre, float* __restrict__ Bim) {
    const int iout = blockIdx.x * blockDim.x + threadIdx.x;
    if (iout >= NTT) return;
    const int itsup = (iout + 1 >= NTT) ? (NTT - 1) : (iout + 1);
    const float dU0 = dUe[iout], dU1 = dUe[itsup];
    const float dV0 = dVe[iout], dV1 = dVe[itsup];
    const float k1sup = Kt1[itsup];

    float Ar = 1.0f, Ai = 0.0f, Br = 0.0f, Bi = 0.0f;
    for (int iin = 0; iin < NSUBI; ++iin) {
        const int it = iout * NSUBI + iin;
        int idx1 = it - 1;
        if (idx1 < 0)        idx1 += NTT;    // negative wrap (it==0 only)
        if (idx1 > NTT - 1)  idx1  = NTT - 1; // OOB clamp
        const float aa  = (float)iin / (float)NSUBI;
        const float om  = 1.0f - aa;
        const float k1  = om * Kt1[idx1] + aa * k1sup;
        const float dUn = om * dU0 + aa * dU1;
        const float dVn = om * dV0 + aa * dV1;
        const float ar = 1.0f - DTF * k1;    // Re(a)
        const float ai = -DTF * FCF;         // Im(a)
        const float br = -DTF * dUn;
        const float bi = -DTF * dVn;
        const float nBr = ar * Br - ai * Bi + br;  // B = a*B + b
        const float nBi = ar * Bi + ai * Br + bi;
        const float nAr = ar * Ar - ai * Ai;       // A = a*A
        const float nAi = ar * Ai + ai * Ar;
        Br = nBr; Bi = nBi; Ar = nAr; Ai = nAi;
    }
    Are[iout] = Ar; Aim[iout] = Ai; Bre[iout] = Br; Bim[iout] = Bi;
}

// ---------------- Kernel 5: parallel scan over 2160 affine maps + output --------
// Single block, 256 threads, 9 chunks/thread. C_hour[j] = B-part of composition
// of chunks 0..j-1 applied to C0=0. out = [Ue+Re(C), Ve+Im(C)].
__global__ void __launch_bounds__(256)
k_scan_out(const float* __restrict__ Are, const float* __restrict__ Aim,
           const float* __restrict__ Bre, const float* __restrict__ Bim,
           const float* __restrict__ Ue,  const float* __restrict__ Ve,
           float* __restrict__ out) {
    const int CH = 9;
    const int t  = threadIdx.x;
    const int beg = t * CH;
    const int end = (beg + CH < NTT) ? (beg + CH) : NTT;

    // local composition of this thread's chunks (later chunk applied after)
    float la_r = 1.0f, la_i = 0.0f, lb_r = 0.0f, lb_i = 0.0f;
    for (int j = beg; j < end; ++j) {
        const float ar = Are[j], ai = Aim[j], br = Bre[j], bi = Bim[j];
        const float na_r = ar * la_r - ai * la_i;
        const float na_i = ar * la_i + ai * la_r;
        const float nb_r = ar * lb_r - ai * lb_i + br;
        const float nb_i = ar * lb_i + ai * lb_r + bi;
        la_r = na_r; la_i = na_i; lb_r = nb_r; lb_i = nb_i;
    }

    __shared__ float sar[256], sai[256], sbr[256], sbi[256];
    sar[t] = la_r; sai[t] = la_i; sbr[t] = lb_r; sbi[t] = lb_i;
    __syncthreads();

    // Hillis-Steele inclusive scan over affine maps: cur = cur ∘ prev
    for (int off = 1; off < 256; off <<= 1) {
        float par = 1.0f, pai = 0.0f, pbr = 0.0f, pbi = 0.0f;
        if (t >= off) { par = sar[t-off]; pai = sai[t-off]; pbr = sbr[t-off]; pbi = sbi[t-off]; }
        const float car = sar[t], cai = sai[t], cbr = sbr[t], cbi = sbi[t];
        __syncthreads();
        sar[t] = car * par - cai * pai;
        sai[t] = car * pai + cai * par;
        sbr[t] = car * pbr - cai * pbi + cbr;
        sbi[t] = car * pbi + cai * pbr + cbi;
        __syncthreads();
    }

    // exclusive prefix for this thread
    float ea_r = 1.0f, ea_i = 0.0f, eb_r = 0.0f, eb_i = 0.0f;
    if (t > 0) { ea_r = sar[t-1]; ea_i = sai[t-1]; eb_r = sbr[t-1]; eb_i = sbi[t-1]; }

    // replay chunks, emitting C_hour[j] (= B-part since C0 = 0) before applying chunk j
    for (int j = beg; j < end; ++j) {
        out[j]       = Ue[j] + eb_r;
        out[NTT + j] = Ve[j] + eb_i;
        const float ar = Are[j], ai = Aim[j], br = Bre[j], bi = Bim[j];
        const float na_r = ar * ea_r - ai * ea_i;
        const float na_i = ar * ea_i + ai * ea_r;
        const float nb_r = ar * eb_r - ai * eb_i + br;
        const float nb_i = ar * eb_i + ai * eb_r + bi;
        ea_r = na_r; ea_i = na_i; eb_r = nb_r; eb_i = nb_i;
    }
}

// ---------------- host launcher ----------------
extern "C" void kernel_launch(void* const* d_in, const int* in_sizes, int n_in,
                              void* d_out, int out_size, void* d_ws, size_t ws_size,
                              hipStream_t stream) {
    (void)in_sizes; (void)n_in; (void)out_size; (void)ws_size;
    const float* pk  = (const float*)d_in[0];   // 18
    const float* TAx = (const float*)d_in[1];   // 2160
    const float* TAy = (const float*)d_in[2];   // 2160
    float* out = (float*)d_out;                 // 2*2160

    float* ws  = (float*)d_ws;
    float* Kt0 = ws + 0 * NTT;
    float* Kt1 = ws + 1 * NTT;
    float* Ue  = ws + 2 * NTT;
    float* Ve  = ws + 3 * NTT;
    float* dUe = ws + 4 * NTT;
    float* dVe = ws + 5 * NTT;
    float* Are = ws + 6 * NTT;
    float* Aim = ws + 7 * NTT;
    float* Bre = ws + 8 * NTT;
    float* Bim = ws + 9 * NTT;

    k_kt_wmma<<<dim3(NTT / 16), dim3(32), 0, stream>>>(pk, Kt0, Kt1);
    k_ekman  <<<dim3((NTT + 255) / 256), dim3(256), 0, stream>>>(Kt0, Kt1, TAx, TAy, Ue, Ve);
    k_grad   <<<dim3((NTT + 255) / 256), dim3(256), 0, stream>>>(Ue, Ve, dUe, dVe);
    k_chunks <<<dim3((NTT + 255) / 256), dim3(256), 0, stream>>>(Kt1, dUe, dVe, Are, Aim, Bre, Bim);
    k_scan_out<<<dim3(1), dim3(256), 0, stream>>>(Are, Aim, Bre, Bim, Ue, Ve, out);
}